// BaselineModel_16209206575815
// MI455X (gfx1250) — compile-verified
//
#include <hip/hip_runtime.h>

#define HID 32

typedef __attribute__((ext_vector_type(2))) float v2f;
typedef __attribute__((ext_vector_type(8))) float v8f;

// ---------------- utility ----------------
__global__ void zero_f32(float* __restrict__ p, int n) {
  for (int i = blockIdx.x * blockDim.x + threadIdx.x; i < n; i += gridDim.x * blockDim.x)
    p[i] = 0.0f;
}

__global__ void scale_copy(float* __restrict__ out, const float* __restrict__ in, float a, int n) {
  for (int i = blockIdx.x * blockDim.x + threadIdx.x; i < n; i += gridDim.x * blockDim.x)
    out[i] = a * in[i];
}

// ---------------- graph normalization ----------------
__global__ void deg_accum(float* __restrict__ deg, const int* __restrict__ src, int e) {
  int i = blockIdx.x * blockDim.x + threadIdx.x;
  if (i < e) atomicAdd(&deg[src[i]], 1.0f);
}

__global__ void deg_to_dinv(float* __restrict__ d, int n) {
  int i = blockIdx.x * blockDim.x + threadIdx.x;
  if (i < n) {
    float v = d[i];
    d[i] = (v > 0.0f) ? rsqrtf(v) : 0.0f;
  }
}

__global__ void edge_norm(float* __restrict__ norm, const int* __restrict__ src,
                          const int* __restrict__ dst, const float* __restrict__ dinv, int e) {
  int i = blockIdx.x * blockDim.x + threadIdx.x;
  if (i < e) norm[i] = -(dinv[src[i]] * dinv[dst[i]]);
}

// ---------------- sparse propagation ----------------
// out[dst] += scale * norm[e] * in[src]   (out pre-initialized: 0 or -T_{k-2})
__global__ void prop_w1(float* __restrict__ out, const float* __restrict__ in,
                        const int* __restrict__ src, const int* __restrict__ dst,
                        const float* __restrict__ norm, float scale, int e) {
  int i = blockIdx.x * blockDim.x + threadIdx.x;
  if (i < e) atomicAdd(&out[dst[i]], scale * norm[i] * in[src[i]]);
}

// one wave per edge: 32 lanes = 32 features (coalesced 128B gather, per-lane atomics)
__global__ void prop_w32(float* __restrict__ out, const float* __restrict__ in,
                         const int* __restrict__ src, const int* __restrict__ dst,
                         const float* __restrict__ norm, float scale, int e) {
  int g = blockIdx.x * blockDim.x + threadIdx.x;
  int edge = g >> 5;
  int lane = g & 31;
  if (edge >= e) return;
  int s = src[edge];
  int d = dst[edge];
  float w = scale * norm[edge];
  atomicAdd(&out[d * HID + lane], w * in[s * HID + lane]);
}

// ---------------- layer 1 dense combine (rank-1 per hop) ----------------
__global__ void cheb1_combine(const float* __restrict__ t0, const float* __restrict__ t1,
                              const float* __restrict__ t2, const float* __restrict__ t3,
                              const float* __restrict__ t4, const float* __restrict__ W1,
                              const float* __restrict__ b1, float* __restrict__ H, int n) {
  int i = blockIdx.x * blockDim.x + threadIdx.x;
  if (i >= n * HID) return;
  int node = i >> 5;
  int j = i & 31;
  float acc = b1[j];
  acc = fmaf(t0[node], W1[0 * HID + j], acc);
  acc = fmaf(t1[node], W1[1 * HID + j], acc);
  acc = fmaf(t2[node], W1[2 * HID + j], acc);
  acc = fmaf(t3[node], W1[3 * HID + j], acc);
  acc = fmaf(t4[node], W1[4 * HID + j], acc);
  H[i] = fmaxf(acc, 0.0f);
}

// ---------------- WMMA dense: H = act( sum_k T_k @ W[k] + b ) ----------------
// Exact-f32 path via V_WMMA_F32_16X16X4_F32.
// Layouts (ISA 7.12.2, wave32): lo = lane&15, hi = lane>>4
//   A 16x4 f32:  comp v of lane L -> (M = lo, K = 2*hi + v)   [chunk offset 4c]
//   B 4x16 f32:  comp v of lane L -> (K = 2*hi + v, N = lo)
//   C/D 16x16:   comp v of lane L -> (M = v + 8*hi, N = lo)
__global__ void cheb_gemm32(const float* __restrict__ T0, const float* __restrict__ T1,
                            const float* __restrict__ T2, const float* __restrict__ T3,
                            const float* __restrict__ T4, const float* __restrict__ W,
                            const float* __restrict__ b, float* __restrict__ H,
                            int doRelu, int n) {
  int wave = (blockIdx.x * blockDim.x + threadIdx.x) >> 5;
  int lane = threadIdx.x & 31;
  int row0 = wave * 16;
  if (row0 >= n) return;  // wave-uniform: EXEC stays all-ones inside active waves
  int lo = lane & 15;
  int hi = lane >> 4;

  v8f c0 = {};  // output cols 0..15
  v8f c1 = {};  // output cols 16..31

#pragma unroll
  for (int k = 0; k < 5; ++k) {
    const float* T  = (k == 0) ? T0 : (k == 1) ? T1 : (k == 2) ? T2 : (k == 3) ? T3 : T4;
    const float* Wk = W + k * HID * HID;
    const float* arow = T + (row0 + lo) * HID;
#pragma unroll
    for (int c = 0; c < 8; ++c) {
      int kk = 4 * c + 2 * hi;
      v2f a = *(const v2f*)(arow + kk);  // consecutive K pair, 8B aligned
      v2f bb0, bb1;
      bb0.x = Wk[kk * HID + lo];
      bb0.y = Wk[(kk + 1) * HID + lo];
      bb1.x = Wk[kk * HID + 16 + lo];
      bb1.y = Wk[(kk + 1) * HID + 16 + lo];
      c0 = __builtin_amdgcn_wmma_f32_16x16x4_f32(false, a, false, bb0, (short)0, c0, false, false);
      c1 = __builtin_amdgcn_wmma_f32_16x16x4_f32(false, a, false, bb1, (short)0, c1, false, false);
    }
  }

  float bias0 = b[lo];
  float bias1 = b[16 + lo];
#pragma unroll
  for (int v = 0; v < 8; ++v) {
    int m = row0 + v + 8 * hi;
    float x0 = c0[v] + bias0;
    float x1 = c1[v] + bias1;
    if (doRelu) { x0 = fmaxf(x0, 0.0f); x1 = fmaxf(x1, 0.0f); }
    H[m * HID + lo]      = x0;
    H[m * HID + 16 + lo] = x1;
  }
}

// ---------------- final linear head ----------------
__global__ void out_bias_init(const float* __restrict__ bl, float* __restrict__ out, int ng) {
  int i = blockIdx.x * blockDim.x + threadIdx.x;
  if (i < ng * 33) out[i] = bl[i % 33];
}

__global__ void final_reduce(const float* __restrict__ H, const float* __restrict__ Wl,
                             float* __restrict__ out, int Lg) {
  int o = blockIdx.y;   // 0..32
  int g = blockIdx.z;   // graph
  int per = (Lg + gridDim.x - 1) / gridDim.x;
  int base = blockIdx.x * per;
  int end = base + per;
  if (end > Lg) end = Lg;
  const float* Hg = H + (size_t)g * Lg;
  float s = 0.0f;
  for (int i = base + threadIdx.x; i < end; i += blockDim.x)
    s += Hg[i] * Wl[(size_t)i * 33 + o];
  __shared__ float sm[256];
  sm[threadIdx.x] = s;
  __syncthreads();
  for (int off = 128; off > 0; off >>= 1) {
    if ((int)threadIdx.x < off) sm[threadIdx.x] += sm[threadIdx.x + off];
    __syncthreads();
  }
  if (threadIdx.x == 0) atomicAdd(&out[g * 33 + o], sm[0]);
}

// ---------------- host ----------------
extern "C" void kernel_launch(void* const* d_in, const int* in_sizes, int n_in,
                              void* d_out, int out_size, void* d_ws, size_t ws_size,
                              hipStream_t stream) {
  (void)n_in; (void)ws_size;
  const float* x  = (const float*)d_in[0];
  const int*   ei = (const int*)d_in[1];
  // d_in[2] = batch (unused; graphs are contiguous blocks)
  const float* W1 = (const float*)d_in[3];
  const float* b1 = (const float*)d_in[4];
  const float* W2 = (const float*)d_in[5];
  const float* b2 = (const float*)d_in[6];
  const float* W3 = (const float*)d_in[7];
  const float* b3 = (const float*)d_in[8];
  const float* Wl = (const float*)d_in[9];
  const float* bl = (const float*)d_in[10];
  float* out = (float*)d_out;

  const int N  = in_sizes[0];
  const int E  = in_sizes[1] / 2;
  const int NG = out_size / 33;
  const int Lg = (N / NG) * HID;

  const int* src = ei;
  const int* dst = ei + E;

  float* ws = (float*)d_ws;
  float* dinv = ws;  ws += N;          // degrees, then d^{-1/2} in place
  float* norm = ws;  ws += E;
  float* t1 = ws;    ws += N;
  float* t2 = ws;    ws += N;
  float* t3 = ws;    ws += N;
  float* t4 = ws;    ws += N;
  const size_t NH = (size_t)N * HID;
  float* h1 = ws;  ws += NH;
  float* P1 = ws;  ws += NH;
  float* P2 = ws;  ws += NH;
  float* P3 = ws;  ws += NH;
  float* P4 = ws;  ws += NH;
  float* h2 = ws;  ws += NH;
  float* h3 = h1;  // h1 is dead after the layer-2 GEMM

  const int B = 256;
  auto nb = [](long long t, int b) { return (int)((t + b - 1) / b); };
  const int NHi = (int)NH;
  const long long EW = (long long)E * 32;

  // ---- normalization ----
  zero_f32<<<nb(N, B), B, 0, stream>>>(dinv, N);
  deg_accum<<<nb(E, B), B, 0, stream>>>(dinv, src, E);
  deg_to_dinv<<<nb(N, B), B, 0, stream>>>(dinv, N);
  edge_norm<<<nb(E, B), B, 0, stream>>>(norm, src, dst, dinv, E);

  // ---- layer 1 (feature width 1) ----
  zero_f32<<<nb(N, B), B, 0, stream>>>(t1, N);
  prop_w1<<<nb(E, B), B, 0, stream>>>(t1, x, src, dst, norm, 1.0f, E);
  scale_copy<<<nb(N, B), B, 0, stream>>>(t2, x, -1.0f, N);
  prop_w1<<<nb(E, B), B, 0, stream>>>(t2, t1, src, dst, norm, 2.0f, E);
  scale_copy<<<nb(N, B), B, 0, stream>>>(t3, t1, -1.0f, N);
  prop_w1<<<nb(E, B), B, 0, stream>>>(t3, t2, src, dst, norm, 2.0f, E);
  scale_copy<<<nb(N, B), B, 0, stream>>>(t4, t2, -1.0f, N);
  prop_w1<<<nb(E, B), B, 0, stream>>>(t4, t3, src, dst, norm, 2.0f, E);
  cheb1_combine<<<nb((long long)N * HID, B), B, 0, stream>>>(x, t1, t2, t3, t4, W1, b1, h1, N);

  const int gemmGrid = nb((long long)(N / 16) * 32, B);

  // ---- layer 2 (width 32) ----
  zero_f32<<<nb(NHi, B), B, 0, stream>>>(P1, NHi);
  prop_w32<<<nb(EW, B), B, 0, stream>>>(P1, h1, src, dst, norm, 1.0f, E);
  scale_copy<<<nb(NHi, B), B, 0, stream>>>(P2, h1, -1.0f, NHi);
  prop_w32<<<nb(EW, B), B, 0, stream>>>(P2, P1, src, dst, norm, 2.0f, E);
  scale_copy<<<nb(NHi, B), B, 0, stream>>>(P3, P1, -1.0f, NHi);
  prop_w32<<<nb(EW, B), B, 0, stream>>>(P3, P2, src, dst, norm, 2.0f, E);
  scale_copy<<<nb(NHi, B), B, 0, stream>>>(P4, P2, -1.0f, NHi);
  prop_w32<<<nb(EW, B), B, 0, stream>>>(P4, P3, src, dst, norm, 2.0f, E);
  cheb_gemm32<<<gemmGrid, B, 0, stream>>>(h1, P1, P2, P3, P4, W2, b2, h2, 1, N);

  // ---- layer 3 (width 32, no relu) ----
  zero_f32<<<nb(NHi, B), B, 0, stream>>>(P1, NHi);
  prop_w32<<<nb(EW, B), B, 0, stream>>>(P1, h2, src, dst, norm, 1.0f, E);
  scale_copy<<<nb(NHi, B), B, 0, stream>>>(P2, h2, -1.0f, NHi);
  prop_w32<<<nb(EW, B), B, 0, stream>>>(P2, P1, src, dst, norm, 2.0f, E);
  scale_copy<<<nb(NHi, B), B, 0, stream>>>(P3, P1, -1.0f, NHi);
  prop_w32<<<nb(EW, B), B, 0, stream>>>(P3, P2, src, dst, norm, 2.0f, E);
  scale_copy<<<nb(NHi, B), B, 0, stream>>>(P4, P2, -1.0f, NHi);
  prop_w32<<<nb(EW, B), B, 0, stream>>>(P4, P3, src, dst, norm, 2.0f, E);
  cheb_gemm32<<<gemmGrid, B, 0, stream>>>(h2, P1, P2, P3, P4, W3, b3, h3, 0, N);

  // ---- head ----
  out_bias_init<<<nb(out_size, B), B, 0, stream>>>(bl, out, NG);
  dim3 fg(32, 33, NG);
  final_reduce<<<fg, 256, 0, stream>>>(h3, Wl, out, Lg);
}